// BoltzmannMachine_84275848282199
// MI455X (gfx1250) — compile-verified
//
#include <hip/hip_runtime.h>
#include <hip/hip_bf16.h>
#include <stdint.h>

typedef __attribute__((ext_vector_type(2))) float v2f;
typedef __attribute__((ext_vector_type(4))) float v4f;
typedef __attribute__((ext_vector_type(8))) float v8f;

#define N_SWEEPS 100
#define MAXN 4096

__device__ __forceinline__ uint32_t hash32(uint32_t x) {
  x ^= x >> 17; x *= 0xed5ad4bbu;
  x ^= x >> 11; x *= 0xac4c1b51u;
  x ^= x >> 15; x *= 0x31848babu;
  x ^= x >> 14;
  return x;
}
__device__ __forceinline__ float uniform01(uint32_t x) {
  return (float)(hash32(x) >> 8) * (1.0f / 16777216.0f);
}

// Kernel 1: vh = clamped ? [v;0] : random +-1 ; build free-bit words and per-chunk "any free" bytes.
__global__ void bm_init_vh(const float* __restrict__ v, const float* __restrict__ clamped,
                           int n, int nv, float* __restrict__ vh,
                           uint32_t* __restrict__ freew, uint8_t* __restrict__ chunkfree) {
  int j = blockIdx.x * blockDim.x + threadIdx.x;
  if (j >= n) return;
  float c = clamped[j];
  float base = (j < nv) ? v[j] : 0.0f;
  float r = (hash32((uint32_t)j * 2654435761u + 12345u) & 1u) ? 1.0f : -1.0f;
  vh[j] = c * base + (1.0f - c) * r;
  int freeb = (c == 0.0f) ? 1 : 0;
  uint32_t ball = (uint32_t)__ballot(freeb);
  int lane = threadIdx.x & 31;
  if (lane == 0) {
    freew[j >> 5] = ball;
    chunkfree[j >> 4]       = (ball & 0xFFFFu) ? 1 : 0;
    chunkfree[(j >> 4) + 1] = (ball >> 16)     ? 1 : 0;
  }
}

// Kernel 2: s = W @ vh, one wave per row, contiguous float4 row reads (L2-resident).
__global__ void bm_init_s(const float* __restrict__ w, const float* __restrict__ vh,
                          int n, float* __restrict__ s) {
  int row  = (int)((blockIdx.x * blockDim.x + threadIdx.x) >> 5);
  int lane = threadIdx.x & 31;
  if (row >= n) return;
  const float* wr = w + (size_t)row * n;
  float acc = 0.0f;
  for (int k = lane * 4; k < n; k += 128) {
    v4f a = *(const v4f*)(wr + k);
    v4f b = *(const v4f*)(vh + k);
    acc += a.x * b.x + a.y * b.y + a.z * b.z + a.w * b.w;
  }
  for (int off = 16; off > 0; off >>= 1) acc += __shfl_down(acc, off, 32);
  if (lane == 0) s[row] = acc;
}

// Kernel 3: sequential Gibbs chain, chunked rank-16 delta updates via V_WMMA_F32_16X16X4_F32.
__global__ void __launch_bounds__(1024, 1) bm_chain(
    const float* __restrict__ w, const float* __restrict__ Tptr,
    const float* __restrict__ vh_g, const float* __restrict__ s_g,
    const uint32_t* __restrict__ freew, const uint8_t* __restrict__ chunkfree,
    int n, float* __restrict__ out) {
  __shared__ float    s_lds[MAXN];
  __shared__ float    vh_lds[MAXN];
  __shared__ float    Blds[256];          // 16x16 diagonal block of W for the chunk
  __shared__ float    delta_lds[16];
  __shared__ uint32_t free_lds[MAXN / 32];
  __shared__ uint8_t  cf_lds[MAXN / 16];
  __shared__ int      flag_lds;

  const int tid  = threadIdx.x;
  const int lane = tid & 31;
  const int wv   = tid >> 5;              // 0..31
  const int nthr = blockDim.x;            // 1024

  for (int i = tid; i < n; i += nthr) { s_lds[i] = s_g[i]; vh_lds[i] = vh_g[i]; }
  for (int i = tid; i < (n >> 5); i += nthr) free_lds[i] = freew[i];
  for (int i = tid; i < (n >> 4); i += nthr) cf_lds[i] = chunkfree[i];
  const float invT = 1.0f / Tptr[0];
  __syncthreads();

  const int nchunks  = n >> 4;
  const int rows_pw  = n >> 5;            // rows of s per wave (128)
  const int m_base   = wv * rows_pw;
  const int nblk     = rows_pw >> 4;      // 16-row blocks per wave (8)
  const int hi       = (lane >> 4) & 1;
  const int r16      = lane & 15;

  for (int sweep = 0; sweep < N_SWEEPS; ++sweep) {
    const uint32_t coff = hash32(0xBADC0DEu + (uint32_t)sweep) % (uint32_t)nchunks;
    for (int ci = 0; ci < nchunks; ++ci) {
      const int c = (int)(((uint32_t)ci * 97u + coff) % (uint32_t)nchunks);
      if (!cf_lds[c]) continue;           // uniform skip: fully clamped chunk
      const int j0 = c << 4;

      // ---- wave 0: resolve the 16 sequential unit decisions ----
      if (wv == 0) {
        // stage 16x16 diagonal block: 2 x float4 per lane (prefetched by wave 1 last chunk)
        {
          int t0 = lane;            // 64 float4 tiles: row = t>>2, col4 = (t&3)*4
          int t1 = lane + 32;
          *(v4f*)&Blds[t0 * 4] =
              *(const v4f*)(w + (size_t)(j0 + (t0 >> 2)) * n + (j0 + ((t0 & 3) << 2)));
          *(v4f*)&Blds[t1 * 4] =
              *(const v4f*)(w + (size_t)(j0 + (t1 >> 2)) * n + (j0 + ((t1 & 3) << 2)));
        }
        float sloc   = s_lds[j0 + r16];
        float vh_old = vh_lds[j0 + r16];
        int anyflip = 0;
        #pragma unroll 1
        for (int k = 0; k < 16; ++k) {
          const int   j    = j0 + k;
          const float s_k  = __shfl(sloc, k, 32);
          const float vh_k = __shfl(vh_old, k, 32);
          const int   fk   = (free_lds[j >> 5] >> (j & 31)) & 1;
          const float u    = uniform01((uint32_t)(sweep * n + j) * 0x9E3779B9u + 7u);
          const float p    = 1.0f / (1.0f + __expf(-s_k * invT));
          const float newv = (u <= p) ? 1.0f : -1.0f;
          const float dk   = fk ? (newv - vh_k) : 0.0f;
          anyflip |= (dk != 0.0f);
          if (lane == k) { delta_lds[k] = dk; if (fk) vh_lds[j] = newv; }
          if (lane < 16 && lane > k) sloc += Blds[lane * 16 + k] * dk;  // in-chunk propagation
        }
        if (lane == 0) flag_lds = anyflip;
      }
      __syncthreads();

      // ---- all 32 waves: s += W[:, j0..j0+15] @ delta via WMMA f32 16x16x4 ----
      if (flag_lds) {
        for (int b = 0; b < nblk; ++b) {
          const int m0 = m_base + (b << 4);
          v8f cacc = {0.f, 0.f, 0.f, 0.f, 0.f, 0.f, 0.f, 0.f};
          #pragma unroll
          for (int k0 = 0; k0 < 16; k0 += 4) {
            // A (16x4): lane L -> row m0+(L&15), K pair (L>=16 ? {2,3} : {0,1}) -> 8B load
            v2f a = *(const v2f*)(w + (size_t)(m0 + r16) * n + (j0 + k0 + hi * 2));
            // B (4x16): delta replicated across all 16 columns
            v2f bb;
            bb.x = delta_lds[k0 + hi * 2];
            bb.y = delta_lds[k0 + hi * 2 + 1];
            cacc = __builtin_amdgcn_wmma_f32_16x16x4_f32(
                false, a, false, bb, (short)0, cacc, false, false);
          }
          // every D column equals the 16-row update; lanes 0 and 16 commit rows 0-7 / 8-15
          if (r16 == 0) {
            const int rb = m0 + hi * 8;
            s_lds[rb + 0] += cacc[0]; s_lds[rb + 1] += cacc[1];
            s_lds[rb + 2] += cacc[2]; s_lds[rb + 3] += cacc[3];
            s_lds[rb + 4] += cacc[4]; s_lds[rb + 5] += cacc[5];
            s_lds[rb + 6] += cacc[6]; s_lds[rb + 7] += cacc[7];
          }
        }
      }

      // ---- wave 1: prefetch next active chunk's diagonal block (hides L2 latency
      //      from the serialized resolve path). Schedule is deterministic & uniform. ----
      if (wv == 1) {
        const int tlin = sweep * nchunks + ci;
        const int tmax = N_SWEEPS * nchunks;
        #pragma unroll 1
        for (int t2 = tlin + 1; t2 < tmax && t2 < tlin + 1 + nchunks; ++t2) {
          const int sw2 = t2 / nchunks;
          const int ci2 = t2 - sw2 * nchunks;
          const uint32_t coff2 = hash32(0xBADC0DEu + (uint32_t)sw2) % (uint32_t)nchunks;
          const int c2 = (int)(((uint32_t)ci2 * 97u + coff2) % (uint32_t)nchunks);
          if (cf_lds[c2]) {
            const int jp = c2 << 4;
            if (lane < 16)
              __builtin_prefetch(w + (size_t)(jp + lane) * n + jp, 0, 3);
            break;
          }
        }
      }
      __syncthreads();
    }
  }
  for (int i = tid; i < n; i += nthr) out[i] = vh_lds[i];
}

extern "C" void kernel_launch(void* const* d_in, const int* in_sizes, int n_in,
                              void* d_out, int out_size, void* d_ws, size_t ws_size,
                              hipStream_t stream) {
  const float* v       = (const float*)d_in[0];
  const float* clamped = (const float*)d_in[1];
  const float* T       = (const float*)d_in[2];
  const float* w       = (const float*)d_in[3];
  const int nv = in_sizes[0];
  const int n  = in_sizes[1];

  float*    vh  = (float*)d_ws;                 // n floats
  float*    s   = vh + n;                       // n floats
  uint32_t* fw  = (uint32_t*)(s + n);           // n/32 words
  uint8_t*  cfr = (uint8_t*)(fw + (n >> 5));    // n/16 bytes

  bm_init_vh<<<(n + 255) / 256, 256, 0, stream>>>(v, clamped, n, nv, vh, fw, cfr);
  bm_init_s<<<n / 8, 256, 0, stream>>>(w, vh, n, s);
  bm_chain<<<1, 1024, 0, stream>>>(w, T, vh, s, fw, cfr, n, (float*)d_out);
}